// _global_attention_sop_triu_22814866277105
// MI455X (gfx1250) — compile-verified
//
#include <hip/hip_runtime.h>
#include <hip/hip_bf16.h>
#include <math.h>

#define NBATCH 64
#define M_NODES 8192
#define DIM 128
#define TRI 8256            // 128*129/2
#define EPS_TRIU 0.001f
#define NS_ITER 5

typedef __attribute__((ext_vector_type(2))) float v2f;
typedef __attribute__((ext_vector_type(8))) float v8f;

// fp32 WMMA: D(16x16,f32) = A(16x4,f32) * B(4x16,f32) + C
// 8-arg form: (neg_a, A, neg_b, B, c_mod, C, reuse_a, reuse_b)
__device__ __forceinline__ v8f wmma4(v2f a, v2f b, v8f c) {
  return __builtin_amdgcn_wmma_f32_16x16x4_f32(false, a, false, b, (short)0, c,
                                               false, false);
}

// ---------------------------------------------------------------------------
// K0: expand W_att (triu-flattened) into symmetric D x D matrix Msym such that
//     x^T Msym x == sum_{i<=j} W[t(i,j)] x_i x_j
// ---------------------------------------------------------------------------
__global__ void k_build_msym(const float* __restrict__ W, float* __restrict__ Msym) {
  int idx = blockIdx.x * 256 + threadIdx.x;
  if (idx >= DIM * DIM) return;
  int i = idx >> 7, j = idx & 127;
  int a = (i < j) ? i : j;
  int bb = (i < j) ? j : i;
  int t = a * DIM - (a * (a - 1)) / 2 + (bb - a);
  float w = W[t];
  Msym[idx] = (i == j) ? w : 0.5f * w;
}

// ---------------------------------------------------------------------------
// K1: s[m] = x_m^T Msym x_m + b     (Y = X @ Msym via fp32 WMMA, then row-dot)
// one block = 16 rows of x, 8 waves cover the 8 column tiles of Y
// ---------------------------------------------------------------------------
__global__ void k_logits(const float* __restrict__ x, const float* __restrict__ Msym,
                         const float* __restrict__ b_att, float* __restrict__ s) {
  __shared__ float lds_s[16];
  int tid  = threadIdx.x;
  int wave = tid >> 5;
  int lane = tid & 31;
  int hf   = lane >> 4;      // which K-pair this lane holds
  int l    = lane & 15;
  int m0   = blockIdx.x * 16;
  int col0 = wave * 16;
  if (tid < 16) lds_s[tid] = 0.f;
  __syncthreads();

  v8f c = {};
  const float* xrow = x + (m0 + l) * DIM;     // A row index = l for both halves
  for (int k = 0; k < DIM; k += 4) {
    int ka = k + 2 * hf;
    v2f a, b;
    a.x = xrow[ka];
    a.y = xrow[ka + 1];
    b.x = Msym[ka * DIM + col0 + l];
    b.y = Msym[(ka + 1) * DIM + col0 + l];
    c = wmma4(a, b, c);
  }
  // partial dot: s[m] += Y[m][col] * x[m][col]
  int col = col0 + l;
#pragma unroll
  for (int r = 0; r < 8; ++r) {
    int row = r + 8 * hf;
    atomicAdd(&lds_s[row], c[r] * x[(m0 + row) * DIM + col]);
  }
  __syncthreads();
  if (tid < 16) s[m0 + tid] = lds_s[tid] + b_att[0];
}

// ---------------------------------------------------------------------------
// K2: segment softmax over sorted batch ids (one block per segment)
// ---------------------------------------------------------------------------
__global__ void k_segsoftmax(const float* __restrict__ s, const int* __restrict__ batch,
                             float* __restrict__ imp) {
  __shared__ float red[256];
  int b = blockIdx.x, tid = threadIdx.x;
  float lm = -3.4e38f;
  for (int m = tid; m < M_NODES; m += 256)
    if (batch[m] == b) lm = fmaxf(lm, s[m]);
  red[tid] = lm;
  __syncthreads();
  for (int off = 128; off > 0; off >>= 1) {
    if (tid < off) red[tid] = fmaxf(red[tid], red[tid + off]);
    __syncthreads();
  }
  float smax = red[0];
  __syncthreads();
  float ls = 0.f;
  for (int m = tid; m < M_NODES; m += 256)
    if (batch[m] == b) ls += __expf(s[m] - smax);
  red[tid] = ls;
  __syncthreads();
  for (int off = 128; off > 0; off >>= 1) {
    if (tid < off) red[tid] += red[tid + off];
    __syncthreads();
  }
  float inv = 1.f / red[0];
  for (int m = tid; m < M_NODES; m += 256)
    if (batch[m] == b) imp[m] = __expf(s[m] - smax) * inv;
}

// ---------------------------------------------------------------------------
// K3: pooled[b] = sum_{m in seg b} imp[m] * x_m x_m^T + eps*I
// WMMA with K = nodes (chunks of 4); weight folded into the A operand.
// One block per batch; wave w owns row-tile w x all 8 col-tiles.
// ---------------------------------------------------------------------------
__global__ void k_pooled(const float* __restrict__ x, const float* __restrict__ imp,
                         const int* __restrict__ batch, float* __restrict__ pooled) {
  __shared__ int seg[2];
  int b = blockIdx.x, tid = threadIdx.x;
  if (tid < 2) {  // lower_bound(batch, b) and lower_bound(batch, b+1)
    int target = b + tid, lo = 0, hi = M_NODES;
    while (lo < hi) { int mid = (lo + hi) >> 1; if (batch[mid] < target) lo = mid + 1; else hi = mid; }
    seg[tid] = lo;
  }
  __syncthreads();
  int start = seg[0], end = seg[1];
  int wave = tid >> 5, lane = tid & 31, hf = lane >> 4, l = lane & 15;
  int ri = wave * 16;

  v8f zero = {};
  v8f acc[8];
#pragma unroll
  for (int t = 0; t < 8; ++t) acc[t] = zero;

  for (int mm = start; mm < end; mm += 4) {
    int n0 = mm + 2 * hf, n1 = n0 + 1;
    bool v0 = n0 < end, v1 = n1 < end;
    float w0 = v0 ? imp[n0] : 0.f;
    float w1 = v1 ? imp[n1] : 0.f;
    v2f a;
    a.x = v0 ? w0 * x[n0 * DIM + ri + l] : 0.f;   // A[i][k] = w_k * x_k[ri+i]
    a.y = v1 ? w1 * x[n1 * DIM + ri + l] : 0.f;
#pragma unroll
    for (int tj = 0; tj < 8; ++tj) {
      v2f bf;
      bf.x = v0 ? x[n0 * DIM + tj * 16 + l] : 0.f; // B[k][j] = x_k[cj+j]
      bf.y = v1 ? x[n1 * DIM + tj * 16 + l] : 0.f;
      acc[tj] = wmma4(a, bf, acc[tj]);
    }
  }
  float* P = pooled + b * DIM * DIM;
#pragma unroll
  for (int tj = 0; tj < 8; ++tj) {
#pragma unroll
    for (int r = 0; r < 8; ++r) {
      int gr = ri + r + 8 * hf;
      int gc = tj * 16 + l;
      float vv = acc[tj][r];
      if (gr == gc) vv += EPS_TRIU;
      P[gr * DIM + gc] = vv;
    }
  }
}

// ---------------------------------------------------------------------------
// K4: Newton-Schulz matrix sqrt per batch, all-WMMA, Y/Z/P in 192KB LDS.
// 512 threads = 16 waves; each wave owns 4 of the 64 16x16 output tiles.
// ---------------------------------------------------------------------------
__global__ void k_sqrtm(const float* __restrict__ pooled, float* __restrict__ out) {
  extern __shared__ float smem[];
  float* Y = smem;                 // 16384 f
  float* Z = smem + DIM * DIM;     // 16384 f
  float* P = smem + 2 * DIM * DIM; // 16384 f
  __shared__ float s_tr;
  int b = blockIdx.x, tid = threadIdx.x;
  const float* A = pooled + b * DIM * DIM;

  if (tid == 0) s_tr = 0.f;
  __syncthreads();
  if (tid < DIM) atomicAdd(&s_tr, A[tid * DIM + tid]);
  __syncthreads();
  float tr = s_tr, invtr = 1.f / tr;

  for (int idx = tid; idx < DIM * DIM; idx += 512) {
    Y[idx] = A[idx] * invtr;
    Z[idx] = ((idx >> 7) == (idx & 127)) ? 1.f : 0.f;
  }
  __syncthreads();

  int wave = tid >> 5, lane = tid & 31, hf = lane >> 4, l = lane & 15;
  v8f zero = {};

  for (int it = 0; it < NS_ITER; ++it) {
    // P = 1.5*I - 0.5 * (Z @ Y)
#pragma unroll
    for (int q = 0; q < 4; ++q) {
      int td = wave * 4 + q, ri = (td >> 3) * 16, cj = (td & 7) * 16;
      v8f c = zero;
      for (int k = 0; k < DIM; k += 4) {
        int ka = k + 2 * hf;
        v2f a, bf;
        a.x = Z[(ri + l) * DIM + ka];  a.y = Z[(ri + l) * DIM + ka + 1];
        bf.x = Y[ka * DIM + cj + l];   bf.y = Y[(ka + 1) * DIM + cj + l];
        c = wmma4(a, bf, c);
      }
#pragma unroll
      for (int r = 0; r < 8; ++r) {
        int gr = ri + r + 8 * hf, gc = cj + l;
        P[gr * DIM + gc] = (gr == gc ? 1.5f : 0.f) - 0.5f * c[r];
      }
    }
    __syncthreads();

    // Ynew = Y @ P (into registers)
    v8f cy[4];
#pragma unroll
    for (int q = 0; q < 4; ++q) {
      int td = wave * 4 + q, ri = (td >> 3) * 16, cj = (td & 7) * 16;
      v8f c = zero;
      for (int k = 0; k < DIM; k += 4) {
        int ka = k + 2 * hf;
        v2f a, bf;
        a.x = Y[(ri + l) * DIM + ka];  a.y = Y[(ri + l) * DIM + ka + 1];
        bf.x = P[ka * DIM + cj + l];   bf.y = P[(ka + 1) * DIM + cj + l];
        c = wmma4(a, bf, c);
      }
      cy[q] = c;
    }
    __syncthreads();  // all reads of Y done

    // commit Ynew; then Znew = P @ Z (Znew never reads Y, so no barrier needed)
#pragma unroll
    for (int q = 0; q < 4; ++q) {
      int td = wave * 4 + q, ri = (td >> 3) * 16, cj = (td & 7) * 16;
#pragma unroll
      for (int r = 0; r < 8; ++r)
        Y[(ri + r + 8 * hf) * DIM + cj + l] = cy[q][r];
    }
    v8f cz[4];
#pragma unroll
    for (int q = 0; q < 4; ++q) {
      int td = wave * 4 + q, ri = (td >> 3) * 16, cj = (td & 7) * 16;
      v8f c = zero;
      for (int k = 0; k < DIM; k += 4) {
        int ka = k + 2 * hf;
        v2f a, bf;
        a.x = P[(ri + l) * DIM + ka];  a.y = P[(ri + l) * DIM + ka + 1];
        bf.x = Z[ka * DIM + cj + l];   bf.y = Z[(ka + 1) * DIM + cj + l];
        c = wmma4(a, bf, c);
      }
      cz[q] = c;
    }
    __syncthreads();  // all reads of Z/P done
#pragma unroll
    for (int q = 0; q < 4; ++q) {
      int td = wave * 4 + q, ri = (td >> 3) * 16, cj = (td & 7) * 16;
#pragma unroll
      for (int r = 0; r < 8; ++r)
        Z[(ri + r + 8 * hf) * DIM + cj + l] = cz[q][r];
    }
    __syncthreads();  // Y & Z committed for next iteration
  }

  // out[b] = triu(Y * sqrt(tr))
  float scale = sqrtf(tr);
  for (int idx = tid; idx < DIM * DIM; idx += 512) {
    int i = idx >> 7, j = idx & 127;
    if (j >= i) {
      int t = i * DIM - (i * (i - 1)) / 2 + (j - i);
      out[b * TRI + t] = Y[idx] * scale;
    }
  }
}

// ---------------------------------------------------------------------------
extern "C" void kernel_launch(void* const* d_in, const int* in_sizes, int n_in,
                              void* d_out, int out_size, void* d_ws, size_t ws_size,
                              hipStream_t stream) {
  const float* x     = (const float*)d_in[0];
  const float* W_att = (const float*)d_in[1];
  const float* b_att = (const float*)d_in[2];
  const int*   batch = (const int*)d_in[3];
  // d_in[4] = edge : unused by the reference computation
  float* out = (float*)d_out;

  char* ws = (char*)d_ws;
  float* Msym   = (float*)(ws);                     // 64 KB
  float* s      = (float*)(ws + 65536);             // 32 KB
  float* imp    = (float*)(ws + 65536 + 32768);     // 32 KB
  float* pooled = (float*)(ws + 131072);            // 4 MB (64 x 128 x 128)

  hipLaunchKernelGGL(k_build_msym, dim3((DIM * DIM + 255) / 256), dim3(256), 0, stream,
                     W_att, Msym);
  hipLaunchKernelGGL(k_logits, dim3(M_NODES / 16), dim3(256), 0, stream,
                     x, Msym, b_att, s);
  hipLaunchKernelGGL(k_segsoftmax, dim3(NBATCH), dim3(256), 0, stream,
                     s, batch, imp);
  hipLaunchKernelGGL(k_pooled, dim3(NBATCH), dim3(256), 0, stream,
                     x, imp, batch, pooled);
  hipLaunchKernelGGL(k_sqrtm, dim3(NBATCH), dim3(512),
                     3 * DIM * DIM * sizeof(float), stream, pooled, out);
}